// GCN_74242804678709
// MI455X (gfx1250) — compile-verified
//
#include <hip/hip_runtime.h>
#include <hip/hip_bf16.h>

// ---------------------------------------------------------------------------
// GCN forward: 2x GCNConv(+ReLU) -> Linear(64->256) -> Linear(256->64)
//              -> log_softmax.   fp32 end-to-end, WMMA f32 16x16x4 GEMMs.
// ---------------------------------------------------------------------------

typedef __attribute__((ext_vector_type(2))) float v2f;
typedef __attribute__((ext_vector_type(8))) float v8f;

#define IN_CH  128
#define H1     64
#define H3     256
#define OUTC   64

// ---------------------------------------------------------------------------
// elementwise helpers
// ---------------------------------------------------------------------------
__global__ void zero_f32(float* __restrict__ p, int n) {
    int i = blockIdx.x * blockDim.x + threadIdx.x;
    if (i < n) p[i] = 0.0f;
}

// deg[dst] += 1 per edge
__global__ void deg_kernel(const long long* __restrict__ ei, float* __restrict__ deg, int E) {
    int e = blockIdx.x * blockDim.x + threadIdx.x;
    if (e < E) {
        int d = (int)ei[E + e];
        unsafeAtomicAdd(&deg[d], 1.0f);
    }
}

// dinv[i] = rsqrt(deg[i] + 1)   (in place over deg buffer)
__global__ void dinv_kernel(float* __restrict__ deg, int n) {
    int i = blockIdx.x * blockDim.x + threadIdx.x;
    if (i < n) deg[i] = rsqrtf(deg[i] + 1.0f);
}

// agg[i][c] = h[i][c] * dinv[i]^2   (self-loop term, 64 channels)
__global__ void selfloop_kernel(const float* __restrict__ h, const float* __restrict__ dinv,
                                float* __restrict__ agg, int n64) {
    int idx = blockIdx.x * blockDim.x + threadIdx.x;
    if (idx < n64) {
        float di = dinv[idx >> 6];
        agg[idx] = h[idx] * di * di;
    }
}

// per (edge, channel): agg[dst][c] += h[src][c] * dinv[src]*dinv[dst]
// 64 consecutive threads handle one edge -> coalesced row read + row atomics.
__global__ void edge_scatter_kernel(const long long* __restrict__ ei,
                                    const float* __restrict__ h,
                                    const float* __restrict__ dinv,
                                    float* __restrict__ agg, int E) {
    int idx = blockIdx.x * blockDim.x + threadIdx.x;
    int e = idx >> 6;
    int c = idx & 63;
    if (e < E) {
        int s = (int)ei[e];
        int d = (int)ei[E + e];
        float nm = dinv[s] * dinv[d];
        unsafeAtomicAdd(&agg[(size_t)d * 64 + c], h[(size_t)s * 64 + c] * nm);
    }
}

// out[i][c] = relu(agg[i][c] + b[c])   (64 channels)
__global__ void bias_relu_kernel(const float* __restrict__ agg, const float* __restrict__ b,
                                 float* __restrict__ out, int n64) {
    int idx = blockIdx.x * blockDim.x + threadIdx.x;
    if (idx < n64) out[idx] = fmaxf(agg[idx] + b[idx & 63], 0.0f);
}

// ---------------------------------------------------------------------------
// WMMA fp32 GEMM:  C[M,N] = A[M,K] @ B[K,N] (+ bias[N])
// One wave computes a 16x64 strip: 4 WMMA 16x16 tiles sharing the A fragment,
// stepping K by 4 (V_WMMA_F32_16X16X4_F32).  M%16==0, N%64==0, K%4==0.
// Guards are wave-uniform => EXEC all ones at each WMMA (ISA requirement).
// ---------------------------------------------------------------------------
__global__ void gemm_wmma_f32(const float* __restrict__ A, const float* __restrict__ B,
                              const float* __restrict__ bias, float* __restrict__ C,
                              int M, int N, int K) {
    const int lane = threadIdx.x & 31;
    const int wave = threadIdx.x >> 5;
    const int nColStrips = N >> 6;
    const int totalStrips = (M >> 4) * nColStrips;
    const int strip = blockIdx.x * (blockDim.x >> 5) + wave;
    if (strip >= totalStrips) return;              // uniform per wave

    const int rowBase = (strip / nColStrips) << 4;
    const int colBase = (strip % nColStrips) << 6;

    const int m  = lane & 15;       // row (A) / col (B,C) within tile
    const int lh = lane >> 4;       // lane half selects K pair {0,1} vs {2,3}

    v8f acc[4];
#pragma unroll
    for (int t = 0; t < 4; ++t) acc[t] = (v8f){0,0,0,0,0,0,0,0};

    const float* arow = A + (size_t)(rowBase + m) * K;

    for (int k = 0; k < K; k += 4) {
        // A fragment 16x4: lanes 0-15 hold K=k..k+1, lanes 16-31 K=k+2..k+3
        v2f a;
        a.x = arow[k + 2 * lh + 0];
        a.y = arow[k + 2 * lh + 1];

        const float* b0 = B + (size_t)(k + 2 * lh + 0) * N + colBase + m;
        const float* b1 = B + (size_t)(k + 2 * lh + 1) * N + colBase + m;

#pragma unroll
        for (int t = 0; t < 4; ++t) {
            v2f bf;
            bf.x = b0[16 * t];
            bf.y = b1[16 * t];
            acc[t] = __builtin_amdgcn_wmma_f32_16x16x4_f32(
                false, a, false, bf, (short)0, acc[t], false, false);
        }
    }

    float bv[4];
#pragma unroll
    for (int t = 0; t < 4; ++t) bv[t] = bias ? bias[colBase + 16 * t + m] : 0.0f;

    // C/D layout: VGPR i -> row rowBase + 8*lh + i, col colBase + 16*t + m
#pragma unroll
    for (int i = 0; i < 8; ++i) {
        const int r = rowBase + 8 * lh + i;
        float* crow = C + (size_t)r * N + colBase + m;
#pragma unroll
        for (int t = 0; t < 4; ++t) crow[16 * t] = acc[t][i] + bv[t];
    }
}

// ---------------------------------------------------------------------------
// log_softmax over 64 channels, one wave per row, lane holds cols c and c+32
// ---------------------------------------------------------------------------
__global__ void log_softmax64_kernel(const float* __restrict__ in, float* __restrict__ out, int n) {
    int row = blockIdx.x * (blockDim.x >> 5) + (threadIdx.x >> 5);
    if (row >= n) return;
    int lane = threadIdx.x & 31;
    const float* r = in + (size_t)row * 64;
    float v0 = r[lane], v1 = r[lane + 32];
    float mx = fmaxf(v0, v1);
#pragma unroll
    for (int off = 16; off; off >>= 1) mx = fmaxf(mx, __shfl_xor(mx, off, 32));
    float s = expf(v0 - mx) + expf(v1 - mx);
#pragma unroll
    for (int off = 16; off; off >>= 1) s += __shfl_xor(s, off, 32);
    float lse = mx + logf(s);
    float* o = out + (size_t)row * 64;
    o[lane]      = v0 - lse;
    o[lane + 32] = v1 - lse;
}

// ---------------------------------------------------------------------------
// launch
// ---------------------------------------------------------------------------
extern "C" void kernel_launch(void* const* d_in, const int* in_sizes, int n_in,
                              void* d_out, int out_size, void* d_ws, size_t ws_size,
                              hipStream_t stream) {
    const float*     x  = (const float*)d_in[0];
    const long long* ei = (const long long*)d_in[1];
    const float* W1 = (const float*)d_in[2]; const float* b1 = (const float*)d_in[3];
    const float* W2 = (const float*)d_in[4]; const float* b2 = (const float*)d_in[5];
    const float* W3 = (const float*)d_in[6]; const float* b3 = (const float*)d_in[7];
    const float* W4 = (const float*)d_in[8]; const float* b4 = (const float*)d_in[9];
    float* out = (float*)d_out;

    const int N = in_sizes[0] / IN_CH;     // 100000
    const int E = in_sizes[1] / 2;         // 1600000

    // workspace layout (floats)
    float* dinv = (float*)d_ws;                       // N (deg -> dinv in place)
    float* bufA = dinv + 100352;                      // N x 64
    float* bufB = bufA + (size_t)N * 64;              // N x 64
    float* bufC = bufB + (size_t)N * 64;              // N x 64
    float* bufD = bufC + (size_t)N * 64;              // N x 256

    const int n64 = N * 64;
    const int TB = 256;
    dim3 blk(TB);

    // degrees -> dinv
    zero_f32<<<(N + TB - 1) / TB, blk, 0, stream>>>(dinv, N);
    deg_kernel<<<(E + TB - 1) / TB, blk, 0, stream>>>(ei, dinv, E);
    dinv_kernel<<<(N + TB - 1) / TB, blk, 0, stream>>>(dinv, N);

    auto gemm = [&](const float* A, const float* B, const float* bias, float* C,
                    int M, int Nn, int K) {
        int strips = (M >> 4) * (Nn >> 6);
        int wavesPerBlock = 128 >> 5;                 // 4 waves
        int grid = (strips + wavesPerBlock - 1) / wavesPerBlock;
        gemm_wmma_f32<<<grid, 128, 0, stream>>>(A, B, bias, C, M, Nn, K);
    };

    const int scatterGrid = (E * 64 + TB - 1) / TB;

    // --- layer 1: GCNConv(128 -> 64) + ReLU ---
    gemm(x, W1, nullptr, bufA, N, H1, IN_CH);                       // h1 = x @ W1
    selfloop_kernel<<<(n64 + TB - 1) / TB, blk, 0, stream>>>(bufA, dinv, bufB, n64);
    edge_scatter_kernel<<<scatterGrid, blk, 0, stream>>>(ei, bufA, dinv, bufB, E);
    bias_relu_kernel<<<(n64 + TB - 1) / TB, blk, 0, stream>>>(bufB, b1, bufA, n64);

    // --- layer 2: GCNConv(64 -> 64) + ReLU ---
    gemm(bufA, W2, nullptr, bufC, N, H1, H1);                       // h2 = x2 @ W2
    selfloop_kernel<<<(n64 + TB - 1) / TB, blk, 0, stream>>>(bufC, dinv, bufB, n64);
    edge_scatter_kernel<<<scatterGrid, blk, 0, stream>>>(ei, bufC, dinv, bufB, E);
    bias_relu_kernel<<<(n64 + TB - 1) / TB, blk, 0, stream>>>(bufB, b2, bufA, n64);

    // --- layer 3: Linear 64 -> 256 (+b3) ---
    gemm(bufA, W3, b3, bufD, N, H3, H1);

    // --- layer 4: Linear 256 -> 64 (+b4) ---
    gemm(bufD, W4, b4, bufC, N, OUTC, H3);

    // --- log_softmax over 64 channels ---
    int rowsPerBlock = TB >> 5;                                     // 8 waves/block
    log_softmax64_kernel<<<(N + rowsPerBlock - 1) / rowsPerBlock, blk, 0, stream>>>(bufC, out, N);
}